// SRMoLELinear_39943195853507
// MI455X (gfx1250) — compile-verified
//
#include <hip/hip_runtime.h>
#include <hip/hip_bf16.h>
#include <stdint.h>

#define DIN 2048
#define DOUT 2048
#define RR 64        // total ranks
#define RH 32        // router rows (R/RP)
#define TOPK 8
#define SCALING 2.0f // 16.0 / 8
#define BK 64        // K-chunk for fused GEMM
#define ROWP 68      // padded LDS row stride (64 data + 4 pad DWORDs from TDM pad)

typedef __attribute__((ext_vector_type(2))) float v2f;
typedef __attribute__((ext_vector_type(8))) float v8f;
typedef __attribute__((ext_vector_type(4))) unsigned int v4u;
typedef __attribute__((ext_vector_type(8))) int v8i;
typedef __attribute__((ext_vector_type(4))) int v4i;

// Launder an LDS byte offset into an SGPR so the compiler cannot fold the
// addrspacecast into a (non-relocatable) static initializer.
__device__ __forceinline__ unsigned int lds_base_off(void* p) {
  unsigned int v = (unsigned int)(uintptr_t)p;   // low 32 bits == LDS offset
  asm volatile("" : "+s"(v));
  return v;
}

// ---------------------------------------------------------------------------
// TDM: build a D# for a 2-D f32 tile load (row-major tensor -> LDS) with
// pad_enable: +4 DWORDs after every 64 DWORDs stored => LDS row stride = 68.
// Wave-level op; call from a single wave only (uniform branch).
// 6-arg builtin: (u32x4 g0, i32x8 g1, i32x4 g2, i32x4 g3, i32x8 extra, i32 cpol)
// ---------------------------------------------------------------------------
__device__ __forceinline__ void tdm_load_tile_f32(
    unsigned int lds_byte_off, const float* gptr,
    unsigned int tile_cols, unsigned int tile_rows,
    unsigned int tensor_cols, unsigned int tensor_rows,
    unsigned long long row_stride_elems)
{
  unsigned long long ga = (unsigned long long)(uintptr_t)gptr;
  v4u g0;
  g0[0] = 1u;                                            // count=1, user mode
  g0[1] = lds_byte_off;                                  // lds_addr
  g0[2] = (unsigned int)ga;                              // global_addr[31:0]
  g0[3] = (unsigned int)((ga >> 32) & 0x01FFFFFFu)       // global_addr[56:32]
        | (2u << 30);                                    // type = 2 (image)
  v8i g1;
  g1[0] = (int)((2u << 16)      // data_size = 4 bytes
              | (1u << 20)      // pad_enable
              | (5u << 22)      // pad_interval: 64 DWORDs
              | (3u << 25));    // pad_amount: 4 DWORDs
  g1[1] = (int)((tensor_cols & 0xFFFFu) << 16);                      // dim0 lo16
  g1[2] = (int)((tensor_cols >> 16) | ((tensor_rows & 0xFFFFu) << 16));
  g1[3] = (int)((tensor_rows >> 16) | (tile_cols << 16));            // tile_dim0
  g1[4] = (int)(tile_rows & 0xFFFFu);                                // tile_dim1
  g1[5] = (int)(unsigned int)(row_stride_elems & 0xFFFFFFFFull);     // stride0
  g1[6] = (int)(unsigned int)((row_stride_elems >> 32) & 0xFFFFull);
  g1[7] = 0;
  v4i gz4 = {0, 0, 0, 0};
  v8i gz8 = {0, 0, 0, 0, 0, 0, 0, 0};
  __builtin_amdgcn_tensor_load_to_lds(g0, g1, gz4, gz4, gz8, 0);
}

// ---------------------------------------------------------------------------
// Kernel 1: router logits -> sigmoid -> repeat x2 + biases -> exact top-8
// (ascending-index strict-max scan == jax top_k tie-break) -> scaled gating.
// One thread per token; router_W (256 KB) lives in L2/L0.
// ---------------------------------------------------------------------------
__global__ __launch_bounds__(256) void router_gating_kernel(
    const float* __restrict__ x, const float* __restrict__ rW,
    const float* __restrict__ biases, float* __restrict__ gs)
{
  __shared__ float lv[256][65];   // 64 logits + pad (conflict-free scans)
  const int tid = threadIdx.x;
  const int n = blockIdx.x * 256 + tid;
  const float* xr = x + (size_t)n * DIN;

  float acc[RH];
#pragma unroll
  for (int r = 0; r < RH; ++r) acc[r] = 0.0f;

  for (int k = 0; k < DIN; k += 4) {
    float4 xv = *(const float4*)(xr + k);
#pragma unroll
    for (int r = 0; r < RH; ++r) {
      float4 wv = *(const float4*)(rW + r * DIN + k);
      acc[r] += xv.x * wv.x + xv.y * wv.y + xv.z * wv.z + xv.w * wv.w;
    }
  }

#pragma unroll
  for (int r = 0; r < RH; ++r) {
    float s = 1.0f / (1.0f + __expf(-acc[r]));
    lv[tid][2 * r]     = s + biases[2 * r];
    lv[tid][2 * r + 1] = s + biases[2 * r + 1];
  }

  float* gr = gs + (size_t)n * RR;
#pragma unroll
  for (int j = 0; j < RR; ++j) gr[j] = 0.0f;

  float vals[TOPK];
  int   idxs[TOPK];
  float sum = 0.0f;
  for (int t = 0; t < TOPK; ++t) {
    float best = -__builtin_inff();
    int bi = 0;
    for (int j = 0; j < RR; ++j) {
      float v = lv[tid][j];
      if (v > best) { best = v; bi = j; }
    }
    lv[tid][bi] = -__builtin_inff();
    vals[t] = best; idxs[t] = bi; sum += best;
  }
  float inv = ((float)TOPK * SCALING) / sum;   // SCALING folded into gating
#pragma unroll
  for (int t = 0; t < TOPK; ++t) gr[idxs[t]] = vals[t] * inv;
}

// ---------------------------------------------------------------------------
// Kernel 2: gmid = (x @ A^T) * gs   via v_wmma_f32_16x16x4_f32.
// 1 wave per 16-token tile; 4 N-tiles cover R=64.
// ---------------------------------------------------------------------------
__global__ __launch_bounds__(128) void lora_mid_kernel(
    const float* __restrict__ x, const float* __restrict__ Amat,
    const float* __restrict__ gs, float* __restrict__ gmid)
{
  const int lane = threadIdx.x & 31;
  const int wave = threadIdx.x >> 5;
  const int lm = lane & 15;
  const int kh = lane >> 4;                  // 0: K=0,1  1: K=2,3
  const int mbase = (blockIdx.x * 4 + wave) * 16;

  v8f acc[4] = {};
  const float* xr = x + (size_t)(mbase + lm) * DIN;

  for (int k0 = 0; k0 < DIN; k0 += 4) {
    const int kk = k0 + 2 * kh;
    v2f a = *(const v2f*)(xr + kk);
#pragma unroll
    for (int t = 0; t < 4; ++t) {
      v2f b = *(const v2f*)(Amat + (size_t)(t * 16 + lm) * DIN + kk);
      acc[t] = __builtin_amdgcn_wmma_f32_16x16x4_f32(
          false, a, false, b, (short)0, acc[t], false, false);
    }
  }

#pragma unroll
  for (int t = 0; t < 4; ++t) {
#pragma unroll
    for (int v = 0; v < 8; ++v) {
      const int m = mbase + v + 8 * kh;
      const int c = t * 16 + lm;
      const size_t o = (size_t)m * RR + c;
      gmid[o] = acc[t][v] * gs[o];
    }
  }
}

// ---------------------------------------------------------------------------
// Kernel 3: out = x @ base_W^T + gmid @ B^T.
// Block = 256 threads (8 waves); tile 128(M) x 64(N); K chunks of 64
// double-buffered in LDS via TDM (wave 0 drives DMA, s_wait_tensorcnt).
// LDS layout: [xs buf0][xs buf1][w buf0][w buf1], all rows padded to 68 f32.
// ---------------------------------------------------------------------------
#define LDSX_SZ (128 * ROWP)   // floats per x buffer
#define LDSW_SZ (64 * ROWP)    // floats per W buffer

__global__ __launch_bounds__(256) void fused_out_kernel(
    const float* __restrict__ x, const float* __restrict__ W,
    const float* __restrict__ gmid, const float* __restrict__ Bw,
    float* __restrict__ out)
{
  __shared__ float smem[2 * LDSX_SZ + 2 * LDSW_SZ];

  const unsigned int base = lds_base_off((void*)smem);   // SGPR, opaque

  const int lane = threadIdx.x & 31;
  const int wave = threadIdx.x >> 5;
  const int lm = lane & 15;
  const int kh = lane >> 4;
  const int mblock = blockIdx.x * 128;
  const int nbase  = blockIdx.y * 64;
  const bool leader = (wave == 0);

  v8f acc[4] = {};

  if (leader) {
    tdm_load_tile_f32(base, x + (size_t)mblock * DIN, BK, 128,
                      DIN, 1u << 20, DIN);
    tdm_load_tile_f32(base + 2u * LDSX_SZ * 4u, W + (size_t)nbase * DIN,
                      BK, 64, DIN, DOUT, DIN);
    __builtin_amdgcn_s_wait_tensorcnt(0);
  }
  __syncthreads();

  const int NKC = DIN / BK;   // 32 chunks
  for (int c = 0; c < NKC; ++c) {
    const unsigned int cur = (unsigned int)(c & 1);
    const unsigned int nxt = cur ^ 1u;
    if (c + 1 < NKC && leader) {
      const int k0 = (c + 1) * BK;
      tdm_load_tile_f32(base + nxt * (LDSX_SZ * 4u),
                        x + (size_t)mblock * DIN + k0, BK, 128,
                        DIN, 1u << 20, DIN);
      tdm_load_tile_f32(base + 2u * LDSX_SZ * 4u + nxt * (LDSW_SZ * 4u),
                        W + (size_t)nbase * DIN + k0, BK, 64,
                        DIN, DOUT, DIN);
    }
    // compute on current buffer
    const float* xb = smem + cur * LDSX_SZ + (size_t)(wave * 16 + lm) * ROWP;
    const float* wb = smem + 2 * LDSX_SZ + cur * LDSW_SZ;
#pragma unroll
    for (int ks = 0; ks < BK; ks += 4) {
      const int kk = ks + 2 * kh;
      v2f a = *(const v2f*)(xb + kk);
#pragma unroll
      for (int t = 0; t < 4; ++t) {
        v2f b = *(const v2f*)(wb + (size_t)(t * 16 + lm) * ROWP + kk);
        acc[t] = __builtin_amdgcn_wmma_f32_16x16x4_f32(
            false, a, false, b, (short)0, acc[t], false, false);
      }
    }
    __syncthreads();                                   // done reading cur
    if (leader) __builtin_amdgcn_s_wait_tensorcnt(0);  // nxt buffer landed
    __syncthreads();                                   // publish
  }

  // LoRA combine: += gmid[tokens, 64] @ Bw[nbase.., 64]^T  (16 WMMA steps)
  {
    const float* gm = gmid + (size_t)(mblock + wave * 16 + lm) * RR;
#pragma unroll
    for (int ks = 0; ks < RR; ks += 4) {
      const int kk = ks + 2 * kh;
      v2f a = *(const v2f*)(gm + kk);
#pragma unroll
      for (int t = 0; t < 4; ++t) {
        v2f b = *(const v2f*)(Bw + (size_t)(nbase + t * 16 + lm) * RR + kk);
        acc[t] = __builtin_amdgcn_wmma_f32_16x16x4_f32(
            false, a, false, b, (short)0, acc[t], false, false);
      }
    }
  }

  // store D tiles: VGPR v -> row v + 8*kh, col lane&15
#pragma unroll
  for (int t = 0; t < 4; ++t) {
#pragma unroll
    for (int v = 0; v < 8; ++v) {
      const int m = mblock + wave * 16 + v + 8 * kh;
      const int col = nbase + t * 16 + lm;
      out[(size_t)m * DOUT + col] = acc[t][v];
    }
  }
}

// ---------------------------------------------------------------------------
extern "C" void kernel_launch(void* const* d_in, const int* in_sizes, int n_in,
                              void* d_out, int out_size, void* d_ws, size_t ws_size,
                              hipStream_t stream) {
  const float* x       = (const float*)d_in[0];
  const float* base_W  = (const float*)d_in[1];
  const float* routerW = (const float*)d_in[2];
  const float* Amat    = (const float*)d_in[3];
  const float* Bw      = (const float*)d_in[4];
  const float* biases  = (const float*)d_in[5];
  float* out = (float*)d_out;

  const int N = in_sizes[0] / DIN;          // 32768 tokens

  float* gs   = (float*)d_ws;               // [N, 64] scaled sparse gating
  float* gmid = gs + (size_t)N * RR;        // [N, 64] gated mid activations

  router_gating_kernel<<<N / 256, 256, 0, stream>>>(x, routerW, biases, gs);
  lora_mid_kernel<<<N / 64, 128, 0, stream>>>(x, Amat, gs, gmid);
  dim3 g3(N / 128, DOUT / 64);
  fused_out_kernel<<<g3, 256, 0, stream>>>(x, base_W, gmid, Bw, out);
}